// SpatialEncoder_26568667693641
// MI455X (gfx1250) — compile-verified
//
#include <hip/hip_runtime.h>
#include <math.h>

typedef __attribute__((ext_vector_type(2))) float v2f;
typedef __attribute__((ext_vector_type(8))) float v8f;

#define C_DIM 64

// Native non-returning f32 atomic add at device scope (L2 RMW).
// Non-returning => tracked on STOREcnt only, no VGPR writeback, max atomic rate.
__device__ __forceinline__ void atomAddF32(float* p, float v) {
  asm volatile("global_atomic_add_f32 %0, %1, off scope:SCOPE_DEV"
               :
               : "v"(p), "v"(v)
               : "memory");
}

// ---------------------------------------------------------------- utilities

__global__ void zero_kernel(float* __restrict__ p, int n) {
  int i = blockIdx.x * blockDim.x + threadIdx.x;
  if (i < n) p[i] = 0.0f;
}

__global__ void deg_kernel(const long long* __restrict__ src,
                           float* __restrict__ deg, int E) {
  int e = blockIdx.x * blockDim.x + threadIdx.x;
  if (e < E) atomAddF32(&deg[(int)src[e]], 1.0f);
}

__global__ void dis_kernel(const float* __restrict__ deg,
                           float* __restrict__ dis, int N) {
  int i = blockIdx.x * blockDim.x + threadIdx.x;
  if (i < N) {
    float d = deg[i];
    dis[i] = (d > 0.0f) ? (1.0f / sqrtf(d)) : 0.0f;
  }
}

// ------------------------------------------------- edge scatter: y += L_hat x
// 16 lanes per edge, each lane handles 4 consecutive channels (float4 gather,
// 4 native f32 atomics). Gathers and atomics land in the 192MB L2.

__global__ __launch_bounds__(256) void scatter_kernel(
    const float* __restrict__ X, const long long* __restrict__ src,
    const long long* __restrict__ dst, const float* __restrict__ dis,
    float* __restrict__ Y, int E) {
  int t = blockIdx.x * blockDim.x + threadIdx.x;
  int e = t >> 4;
  if (e >= E) return;
  int c4 = (t & 15) << 2;
  int s = (int)src[e];
  int d = (int)dst[e];
  float w = -(dis[s] * dis[d]);
  const float4 v = *(const float4*)(X + (size_t)s * C_DIM + c4);
  float* yp = Y + (size_t)d * C_DIM + c4;
  atomAddF32(yp + 0, w * v.x);
  atomAddF32(yp + 1, w * v.y);
  atomAddF32(yp + 2, w * v.z);
  atomAddF32(yp + 3, w * v.w);
}

// --------------------------------------- fused dual-GEMM: Out = X*W0 + Y*W1 + b
// fp32 WMMA 16x16x4. One 16x16 tile per wave; 4 waves/block cover N=64.
// A layout: lane%16 = M, K = k + 2*(lane/16) + {0,1}.
// B layout: lane%16 = N, K = k + 2*(lane/16) + {0,1}.
// C/D layout: VGPR r, lane l -> M = r + 8*(l/16), N = l%16.

__global__ __launch_bounds__(128) void cheb_gemm_kernel(
    const float* __restrict__ X, const float* __restrict__ Y,
    const float* __restrict__ W0, const float* __restrict__ W1,
    const float* __restrict__ bias, float* __restrict__ Out, int Nnodes,
    int do_relu) {
  const int row0 = blockIdx.x * 16;
  const int wave = threadIdx.x >> 5;  // 0..3 -> N tile
  const int lane = threadIdx.x & 31;
  const int n0 = wave * 16;
  const int lm = lane & 15;   // M for A, N for B/C
  const int lh = lane >> 4;   // K-pair select / M-half select

  // Accumulator seeded with bias (same value down each column).
  v8f c;
  float bv = bias[n0 + lm];
#pragma unroll
  for (int r = 0; r < 8; ++r) c[r] = bv;

  // Clamp the A-row so a non-multiple-of-16 node count would still load
  // safely (N_NODES=50000 is an exact multiple of 16, so this is a no-op).
  const int rowA = (row0 + lm < Nnodes) ? (row0 + lm) : (Nnodes - 1);
  const float* xrow = X + (size_t)rowA * C_DIM + 2 * lh;
  const float* yrow = Y + (size_t)rowA * C_DIM + 2 * lh;
  const float* w0c = W0 + (size_t)(2 * lh) * C_DIM + n0 + lm;
  const float* w1c = W1 + (size_t)(2 * lh) * C_DIM + n0 + lm;

#pragma unroll
  for (int k = 0; k < C_DIM; k += 4) {
    v2f a, b;
    a.x = xrow[k];
    a.y = xrow[k + 1];
    b.x = w0c[(size_t)k * C_DIM];
    b.y = w0c[(size_t)(k + 1) * C_DIM];
    c = __builtin_amdgcn_wmma_f32_16x16x4_f32(false, a, false, b, (short)0, c,
                                              false, false);
  }
#pragma unroll
  for (int k = 0; k < C_DIM; k += 4) {
    v2f a, b;
    a.x = yrow[k];
    a.y = yrow[k + 1];
    b.x = w1c[(size_t)k * C_DIM];
    b.y = w1c[(size_t)(k + 1) * C_DIM];
    c = __builtin_amdgcn_wmma_f32_16x16x4_f32(false, a, false, b, (short)0, c,
                                              false, false);
  }

#pragma unroll
  for (int r = 0; r < 8; ++r) {
    int m = row0 + r + 8 * lh;
    if (m < Nnodes) {
      float v = c[r];
      if (do_relu) v = fmaxf(v, 0.0f);
      Out[(size_t)m * C_DIM + n0 + lm] = v;
    }
  }
}

// ---------------------------------------------------------------- launcher

extern "C" void kernel_launch(void* const* d_in, const int* in_sizes, int n_in,
                              void* d_out, int out_size, void* d_ws,
                              size_t ws_size, hipStream_t stream) {
  const float* x = (const float*)d_in[0];
  const long long* ei = (const long long*)d_in[1];  // int64 (2, E)
  const float* W1_0 = (const float*)d_in[2];
  const float* W1_1 = (const float*)d_in[3];
  const float* b1 = (const float*)d_in[4];
  const float* W2_0 = (const float*)d_in[5];
  const float* W2_1 = (const float*)d_in[6];
  const float* b2 = (const float*)d_in[7];
  float* out = (float*)d_out;

  const int N = in_sizes[0] / C_DIM;  // 50000
  const int E = in_sizes[1] / 2;      // 800000
  const long long* src = ei;
  const long long* dst = ei + E;

  // Workspace layout (floats): deg[N] | dis[N] | y[N*64] | h[N*64]  (~26 MB)
  float* deg = (float*)d_ws;
  float* dis = deg + N;
  float* y = dis + N;
  float* h = y + (size_t)N * C_DIM;

  const int NB = 256;
  const int nfeat = N * C_DIM;

  // degree + normalization (edge weights shared by both layers)
  zero_kernel<<<(N + NB - 1) / NB, NB, 0, stream>>>(deg, N);
  deg_kernel<<<(E + NB - 1) / NB, NB, 0, stream>>>(src, deg, E);
  dis_kernel<<<(N + NB - 1) / NB, NB, 0, stream>>>(deg, dis, N);

  const int mtiles = (N + 15) / 16;
  const int sgrid = ((E * 16) + NB - 1) / NB;

  // Layer 1: y = L_hat x ; h = relu(x W1_0 + y W1_1 + b1)
  zero_kernel<<<(nfeat + NB - 1) / NB, NB, 0, stream>>>(y, nfeat);
  scatter_kernel<<<sgrid, NB, 0, stream>>>(x, src, dst, dis, y, E);
  cheb_gemm_kernel<<<mtiles, 128, 0, stream>>>(x, y, W1_0, W1_1, b1, h, N, 1);

  // Layer 2: y = L_hat h ; out = h W2_0 + y W2_1 + b2
  zero_kernel<<<(nfeat + NB - 1) / NB, NB, 0, stream>>>(y, nfeat);
  scatter_kernel<<<sgrid, NB, 0, stream>>>(h, src, dst, dis, y, E);
  cheb_gemm_kernel<<<mtiles, 128, 0, stream>>>(h, y, W2_0, W2_1, b2, out, N, 0);
}